// WellTemparedMetaDynamics_36000415875082
// MI455X (gfx1250) — compile-verified
//
#include <hip/hip_runtime.h>
#include <hip/hip_bf16.h>

// Well-tempered metadynamics hill append:
//   eng[b] = sum_h hgt_h[b,h] * exp(-0.5 * sum_d (cen-col)^2 * prc_h)
//   new_hgt[b] = hgt * exp(-eng[b] / ((gam-1)*kbt[b]))
//   outputs: cen_out=(B,H+1,D) concat, prc_out=(B,H+1,D) concat, hgt_out=(B,H+1)
//
// Memory-bound (~870 MB @ 23.3 TB/s ~= 37 us). Strategy: single fused streaming
// pass (read cen/prc once, write concat bodies with NT b128 stores, reduce eng
// per block deterministically into d_ws), then a tiny finalize kernel.
// hgt_h stream goes through the CDNA5 async global<->LDS path.

#define Bc 32
#define Hc 200000
#define Dc 8
#define HD4 (Hc * Dc / 4)                 // 400000 float4 per batch row
#define TPB 256
#define NCHUNK ((HD4 + TPB - 1) / TPB)    // 1563 blocks per batch row

typedef float v4f __attribute__((ext_vector_type(4)));

__global__ __launch_bounds__(TPB) void wtmd_main(
    const float* __restrict__ col,     // (B,D)
    const float* __restrict__ cen,     // (B,H,D)
    const float* __restrict__ prc_h,   // (B,H,D)
    const float* __restrict__ hgt_h,   // (B,H)
    float* __restrict__ cen_out,       // (B,H+1,D)
    float* __restrict__ prc_out,       // (B,H+1,D)
    float* __restrict__ hgt_out,       // (B,H+1)
    float* __restrict__ ws)            // (B, NCHUNK) partial eng sums
{
    __shared__ float shgt[TPB];        // async-load staging for hgt_h
    __shared__ float wred[TPB / 32];   // per-wave partials (wave32!)

    const int tid   = threadIdx.x;
    const int b     = blockIdx.y;
    const int chunk = blockIdx.x;
    const int f4    = chunk * TPB + tid;       // float4 index within row b
    const bool active = (f4 < HD4);

    const int hill = f4 >> 1;                  // hill index (pair of lanes / hill)
    const int half = f4 & 1;                   // which 4 of the 8 dims

    // ---- CDNA5 async path: hgt_h[b,hill] --> this lane's LDS slot (ASYNCcnt-tracked) ----
    const float* hbase = hgt_h + (size_t)b * Hc;              // uniform -> SGPR pair
    const unsigned lds_off = (unsigned)(size_t)&shgt[tid];    // LDS byte offset
    const unsigned hoff    = (unsigned)hill * 4u;             // per-lane byte offset
    if (active) {
        asm volatile("global_load_async_to_lds_b32 %0, %1, %2 offset:0"
                     :: "v"(lds_off), "v"(hoff), "s"(hbase) : "memory");
    }

    // ---- main streams: NT b128 load, compute, NT b128 store (concat body) ----
    float partial = 0.0f;
    if (active) {
        const size_t inoff  = (size_t)b * (Hc * Dc)       + (size_t)f4 * 4;
        const size_t outoff = (size_t)b * ((Hc + 1) * Dc) + (size_t)f4 * 4;
        v4f c = __builtin_nontemporal_load((const v4f*)(cen   + inoff));
        v4f p = __builtin_nontemporal_load((const v4f*)(prc_h + inoff));
        v4f q = *(const v4f*)(col + b * Dc + half * 4);   // broadcast, L2-resident
        v4f r = c - q;
        partial = r.x * r.x * p.x + r.y * r.y * p.y
                + r.z * r.z * p.z + r.w * r.w * p.w;
        __builtin_nontemporal_store(c, (v4f*)(cen_out + outoff));
        __builtin_nontemporal_store(p, (v4f*)(prc_out + outoff));
    }

    // pair exchange: lanes (2k,2k+1) hold the two halves of hill k (wave32 shfl)
    float tot = partial + __shfl_xor(partial, 1);

    // wait for async hgt load, read it back from LDS
    asm volatile("s_wait_asynccnt 0x0" ::: "memory");
    const float hv = shgt[tid];

    float e = 0.0f;
    if (active && half == 0) {
        e = hv * __expf(-0.5f * tot);
        // DMA the staged hgt value LDS --> hgt_out (concat body), no VGPR data
        float* obase = hgt_out + (size_t)b * (Hc + 1);
        asm volatile("global_store_async_from_lds_b32 %0, %1, %2 offset:0"
                     :: "v"(hoff), "v"(lds_off), "s"(obase) : "memory");
    }

    // deterministic wave32 tree reduction of e
    #pragma unroll
    for (int off = 16; off >= 1; off >>= 1)
        e += __shfl_xor(e, off);

    const int wid = tid >> 5;
    if ((tid & 31) == 0) wred[wid] = e;
    __syncthreads();
    if (tid == 0) {
        float s = 0.0f;
        #pragma unroll
        for (int w = 0; w < TPB / 32; ++w) s += wred[w];
        ws[(size_t)b * NCHUNK + chunk] = s;   // written every call: deterministic
    }
    // s_endpgm performs an implicit wait-idle, covering in-flight async stores.
}

__global__ __launch_bounds__(256) void wtmd_final(
    const float* __restrict__ col,   // (B,D)
    const float* __restrict__ kbt,   // (B,)
    const float* __restrict__ prc,   // (D,)
    const float* __restrict__ hgt,   // scalar
    const float* __restrict__ gam,   // scalar
    const float* __restrict__ ws,    // (B, NCHUNK)
    float* __restrict__ cen_out,
    float* __restrict__ prc_out,
    float* __restrict__ hgt_out)
{
    __shared__ float red[256];
    const int b = blockIdx.x, tid = threadIdx.x;

    // fixed-order strided partial sum, then fixed-pair tree: deterministic
    float s = 0.0f;
    for (int i = tid; i < NCHUNK; i += 256)
        s += ws[(size_t)b * NCHUNK + i];
    red[tid] = s;
    __syncthreads();
    for (int stride = 128; stride >= 1; stride >>= 1) {
        if (tid < stride) red[tid] += red[tid + stride];
        __syncthreads();
    }

    if (tid < Dc) {
        const size_t tail = ((size_t)b * (Hc + 1) + Hc) * Dc + tid;
        cen_out[tail] = col[b * Dc + tid];
        prc_out[tail] = prc[tid];
    }
    if (tid == 0) {
        const float det = (gam[0] - 1.0f) * kbt[b];       // gam*kbt - kbt
        const float nh  = hgt[0] * __expf(-red[0] / det);
        hgt_out[(size_t)b * (Hc + 1) + Hc] = nh;
    }
}

extern "C" void kernel_launch(void* const* d_in, const int* in_sizes, int n_in,
                              void* d_out, int out_size, void* d_ws, size_t ws_size,
                              hipStream_t stream) {
    const float* col   = (const float*)d_in[0];
    const float* cen   = (const float*)d_in[1];
    const float* prc_h = (const float*)d_in[2];
    const float* hgt_h = (const float*)d_in[3];
    const float* kbt   = (const float*)d_in[4];
    const float* prc   = (const float*)d_in[5];
    const float* hgt   = (const float*)d_in[6];
    const float* gam   = (const float*)d_in[7];

    float* out = (float*)d_out;
    const size_t nOut1 = (size_t)Bc * (Hc + 1) * Dc;   // 51,200,256
    float* cen_out = out;
    float* prc_out = out + nOut1;
    float* hgt_out = out + 2 * nOut1;
    float* ws = (float*)d_ws;                          // needs 32*1563*4 = ~200 KB

    dim3 grid(NCHUNK, Bc);
    hipLaunchKernelGGL(wtmd_main, grid, dim3(TPB), 0, stream,
                       col, cen, prc_h, hgt_h, cen_out, prc_out, hgt_out, ws);
    hipLaunchKernelGGL(wtmd_final, dim3(Bc), dim3(256), 0, stream,
                       col, kbt, prc, hgt, gam, ws, cen_out, prc_out, hgt_out);
}